// JKNet_15779709846033
// MI455X (gfx1250) — compile-verified
//
#include <hip/hip_runtime.h>
#include <math.h>

typedef float v2f __attribute__((ext_vector_type(2)));
typedef float v8f __attribute__((ext_vector_type(8)));

#define DD 128
#define CDIM 40

// ---------------- utility kernels ----------------

__global__ __launch_bounds__(256) void zero_kernel(float* __restrict__ p, int n) {
  int i = blockIdx.x * 256 + threadIdx.x;
  if (i < n) p[i] = 0.0f;
}

__global__ __launch_bounds__(256) void deg_kernel(const long long* __restrict__ dst,
                                                  float* __restrict__ deg, int E) {
  int e = blockIdx.x * 256 + threadIdx.x;
  if (e < E) atomicAdd(&deg[(int)dst[e]], 1.0f);
}

__global__ __launch_bounds__(256) void dinv_kernel(float* __restrict__ deg, int N) {
  int i = blockIdx.x * 256 + threadIdx.x;
  if (i < N) deg[i] = rsqrtf(deg[i] + 1.0f);  // deg counts A, +1 for self-loop
}

// ---------------- fp32 WMMA GEMM: out[N x 128] = A[N x 128] @ W[128 x 128] ----------------
// Block = 256 threads = 8 waves. Block tile = 16 rows x 128 cols (wave w owns cols 16w..16w+15).
// A tile staged in LDS (8 KB) and reused by all 8 waves. K stepped by 4 via V_WMMA_F32_16X16X4_F32.

__global__ __launch_bounds__(256) void gemm_wmma_kernel(
    const float* __restrict__ A, const float* __restrict__ W,
    const float* __restrict__ bias, float* __restrict__ out,
    int N, int fuse_bias_relu)
{
  __shared__ float As[16][DD + 4];
  const int row0 = blockIdx.x * 16;

  for (int t = threadIdx.x; t < 16 * DD; t += 256) {
    const int r = t >> 7, c = t & (DD - 1);
    const int gr = row0 + r;
    As[r][c] = (gr < N) ? A[(size_t)gr * DD + c] : 0.0f;
  }
  __syncthreads();

  const int wave = threadIdx.x >> 5;
  const int lane = threadIdx.x & 31;
  const int half = lane >> 4;   // 0: lanes 0-15, 1: lanes 16-31
  const int l15  = lane & 15;
  const int col  = wave * 16 + l15;

  v8f acc = {};
  for (int k = 0; k < DD; k += 4) {
    const int kb = k + 2 * half;
    v2f a, b;
    // A 16x4 fragment: row = l15; VGPR0/1 hold K = kb, kb+1
    a.x = As[l15][kb];
    a.y = As[l15][kb + 1];
    // B 4x16 fragment: col = wave*16 + l15; VGPR0/1 hold K-rows kb, kb+1
    b.x = W[(size_t)(kb + 0) * DD + col];
    b.y = W[(size_t)(kb + 1) * DD + col];
    acc = __builtin_amdgcn_wmma_f32_16x16x4_f32(
        /*neg_a=*/false, a, /*neg_b=*/false, b,
        /*c_mod=*/(short)0, acc, /*reuse_a=*/false, /*reuse_b=*/false);
  }

#pragma unroll
  for (int v = 0; v < 8; ++v) {
    const int r = row0 + v + 8 * half;   // D tile: row = v + 8*half
    if (r < N) {
      float val = acc[v];
      if (fuse_bias_relu) { val += bias[col]; val = fmaxf(val, 0.0f); }
      out[(size_t)r * DD + col] = val;
    }
  }
}

// ---------------- edge scatter: acc[dst] += hw[src] * dinv[src]*dinv[dst] ----------------
// One wave per edge; float4 gather per lane (GLOBAL_LOAD_B128), 4 float atomics per lane.

__global__ __launch_bounds__(256) void edge_agg_kernel(
    const float* __restrict__ hw, const long long* __restrict__ src,
    const long long* __restrict__ dst, const float* __restrict__ dinv,
    float* __restrict__ acc, int E)
{
  const int e = blockIdx.x * 8 + (threadIdx.x >> 5);
  if (e >= E) return;
  const int lane = threadIdx.x & 31;
  const int s = (int)src[e];
  const int d = (int)dst[e];
  const float w = dinv[s] * dinv[d];

  const float4 v = ((const float4*)(hw + (size_t)s * DD))[lane];
  float* p = acc + (size_t)d * DD + lane * 4;
  atomicAdd(p + 0, v.x * w);
  atomicAdd(p + 1, v.y * w);
  atomicAdd(p + 2, v.z * w);
  atomicAdd(p + 3, v.w * w);
}

// ---------------- fused self-loop + bias + BN(eval) + ReLU + JK-max ----------------

__global__ __launch_bounds__(256) void bn_relu_jk_kernel(
    const float* __restrict__ acc, const float* __restrict__ hw,
    const float* __restrict__ dinv, const float* __restrict__ bias,
    const float* __restrict__ gamma, const float* __restrict__ beta,
    const float* __restrict__ mean, const float* __restrict__ var,
    float* __restrict__ h, float* __restrict__ jk, int ND)
{
  const int i = blockIdx.x * 256 + threadIdx.x;
  if (i >= ND) return;
  const int n = i >> 7, c = i & (DD - 1);
  float sn = dinv[n]; sn *= sn;                       // self-loop weight dinv^2
  float val = acc[i] + hw[i] * sn + bias[c];
  val = (val - mean[c]) * rsqrtf(var[c] + 1e-5f) * gamma[c] + beta[c];
  val = fmaxf(val, 0.0f);                             // ReLU
  h[i] = val;
  jk[i] = fmaxf(jk[i], val);                          // JumpingKnowledge 'max'
}

// ---------------- head: logits = h @ lin2_w + b2 ; log_softmax ----------------
// One wave per node; lane l owns columns l and l+32 (valid for l<8).

__global__ __launch_bounds__(256) void head_kernel(
    const float* __restrict__ h, const float* __restrict__ w2,
    const float* __restrict__ b2, float* __restrict__ out, int N)
{
  const int node = blockIdx.x * 8 + (threadIdx.x >> 5);
  if (node >= N) return;
  const int lane = threadIdx.x & 31;
  const float* hr = h + (size_t)node * DD;
  const int c0 = lane, c1 = lane + 32;
  float v0 = 0.0f, v1 = 0.0f;
  for (int d = 0; d < DD; ++d) {
    const float hv = hr[d];
    v0 += hv * w2[d * CDIM + c0];
    if (c1 < CDIM) v1 += hv * w2[d * CDIM + c1];
  }
  v0 += b2[c0];
  v1 = (c1 < CDIM) ? (v1 + b2[c1]) : -INFINITY;

  float m = fmaxf(v0, v1);
  for (int off = 16; off; off >>= 1) m = fmaxf(m, __shfl_xor(m, off, 32));
  float s = expf(v0 - m) + ((c1 < CDIM) ? expf(v1 - m) : 0.0f);
  for (int off = 16; off; off >>= 1) s += __shfl_xor(s, off, 32);
  const float lse = logf(s);

  out[(size_t)node * CDIM + c0] = v0 - m - lse;
  if (c1 < CDIM) out[(size_t)node * CDIM + c1] = v1 - m - lse;
}

// ---------------- launch ----------------

extern "C" void kernel_launch(void* const* d_in, const int* in_sizes, int n_in,
                              void* d_out, int out_size, void* d_ws, size_t ws_size,
                              hipStream_t stream) {
  const float*     x        = (const float*)d_in[0];
  const long long* edge     = (const long long*)d_in[1];
  const float*     conv_w   = (const float*)d_in[2];
  const float*     conv_b   = (const float*)d_in[3];
  const float*     bn_gamma = (const float*)d_in[4];
  const float*     bn_beta  = (const float*)d_in[5];
  const float*     bn_mean  = (const float*)d_in[6];
  const float*     bn_var   = (const float*)d_in[7];
  const float*     lin1_w   = (const float*)d_in[8];
  const float*     lin1_b   = (const float*)d_in[9];
  const float*     lin2_w   = (const float*)d_in[10];
  const float*     lin2_b   = (const float*)d_in[11];
  float* out = (float*)d_out;

  const int N = in_sizes[0] / DD;        // 50000
  const int E = in_sizes[1] / 2;         // 800000
  const int L = in_sizes[2] / (DD * DD); // 4

  const long long* src = edge;
  const long long* dst = edge + E;

  // workspace carve-up (floats): dinv[N] | h[N*D] | hw[N*D] | acc[N*D] | jk[N*D]
  float* ws   = (float*)d_ws;
  float* dinv = ws;
  float* h    = dinv + ((N + 63) & ~63);
  float* hw   = h   + (size_t)N * DD;
  float* acc  = hw  + (size_t)N * DD;
  float* jk   = acc + (size_t)N * DD;

  const int ND = N * DD;
  dim3 b256(256);

  zero_kernel<<<(N + 255) / 256, b256, 0, stream>>>(dinv, N);
  deg_kernel<<<(E + 255) / 256, b256, 0, stream>>>(dst, dinv, E);
  dinv_kernel<<<(N + 255) / 256, b256, 0, stream>>>(dinv, N);
  zero_kernel<<<(ND + 255) / 256, b256, 0, stream>>>(jk, ND);

  const int gemm_blocks = (N + 15) / 16;
  for (int i = 0; i < L; ++i) {
    const float* hin = (i == 0) ? x : h;
    gemm_wmma_kernel<<<gemm_blocks, b256, 0, stream>>>(
        hin, conv_w + (size_t)i * DD * DD, nullptr, hw, N, 0);
    zero_kernel<<<(ND + 255) / 256, b256, 0, stream>>>(acc, ND);
    edge_agg_kernel<<<(E + 7) / 8, b256, 0, stream>>>(hw, src, dst, dinv, acc, E);
    bn_relu_jk_kernel<<<(ND + 255) / 256, b256, 0, stream>>>(
        acc, hw, dinv,
        conv_b   + (size_t)i * DD, bn_gamma + (size_t)i * DD,
        bn_beta  + (size_t)i * DD, bn_mean  + (size_t)i * DD,
        bn_var   + (size_t)i * DD, h, jk, ND);
  }

  // MLP head: lin1 (WMMA, fused bias+ReLU), then lin2 + log_softmax
  gemm_wmma_kernel<<<gemm_blocks, b256, 0, stream>>>(jk, lin1_w, lin1_b, hw, N, 1);
  head_kernel<<<(N + 7) / 8, b256, 0, stream>>>(hw, lin2_w, lin2_b, out, N);
}